// RetrieveAndReadFramework_37151467110402
// MI455X (gfx1250) — compile-verified
//
#include <hip/hip_runtime.h>
#include <hip/hip_bf16.h>

#define DDIM    128
#define NLAYERS 5
#define NNODES  50000
#define NEDGES  800000
#define BATCHSZ 1024
#define NENT    50000

typedef __attribute__((ext_vector_type(2))) float v2f;
typedef __attribute__((ext_vector_type(4))) float v4f;
typedef __attribute__((ext_vector_type(8))) float v8f;

// ---------------------------------------------------------------------------
// SpMM scatter: agg[row] += val * x[col].  One wave32 per edge; each lane
// owns 4 contiguous dims (32 lanes * 4 = 128).  Unsorted COO forces a
// scatter; f32 global atomics lower to global_atomic_add_f32 (L2 RMW).
// Gather working set (25.6 MB) is L2-resident on MI455X (192 MB L2).
// ---------------------------------------------------------------------------
__global__ void spmm_edge_kernel(const float* __restrict__ x,
                                 const float* __restrict__ edge_val,
                                 const int*   __restrict__ edge_row,
                                 const int*   __restrict__ edge_col,
                                 float*       __restrict__ agg,
                                 int n_edges) {
  int wave = (int)((blockIdx.x * blockDim.x + threadIdx.x) >> 5);
  int lane = (int)(threadIdx.x & 31);
  if (wave >= n_edges) return;

  int   row = edge_row[wave];
  int   col = edge_col[wave];
  float val = edge_val[wave];

  const v4f xv = *(const v4f*)(x + (size_t)col * DDIM + lane * 4);
  float* dst = agg + (size_t)row * DDIM + lane * 4;
#pragma unroll
  for (int j = 0; j < 4; ++j) {
    __hip_atomic_fetch_add(dst + j, val * xv[j],
                           __ATOMIC_RELAXED, __HIP_MEMORY_SCOPE_AGENT);
  }
}

// ---------------------------------------------------------------------------
// fp32 WMMA GEMM with LDS-staged B panel, per-wave register blocking:
//   C[M,N] = act(A[M,K] @ B[K,N] + bias[N])
//
// Block = 256 threads (8 waves). Each block stages an NB = NTN*16 column
// panel of B into LDS (pair-interleaved along K so a lane's WMMA B-fragment
// (B[k][n], B[k+1][n]) is one aligned ds_load_b64; the compiler fuses pairs
// into ds_load_2addr_b64). Each wave computes MT M-tiles x NTN N-tiles:
// per k-step MT A loads + NTN LDS B loads -> MT*NTN WMMAs. The WMMA chain is
// ordered t-outer / mi-inner so both consumers of a B pair are adjacent.
//
// __launch_bounds__(256, 1): the MT=2 accumulator set (16 x v8f = 128 VGPRs)
// must stay resident; the default occupancy heuristic otherwise caps the
// VGPR budget and spills accumulators to scratch inside the inner loop.
//
// Tail handling (mtiles not divisible by MT): loads clamp the tile index
// (garbage accumulated but never stored); stores are guarded per M-tile.
// All guards are wave-uniform -> EXEC all-ones for every WMMA.
//
// Fragment layouts (CDNA5 ISA 7.12.2, 32-bit data):
//   A 16x4 :  lanes 0-15 -> row m0+lane, K={k0,k0+1}; lanes 16-31 -> K={k0+2,k0+3}
//   B 4x16 :  VGPR0/1 = K pair per lane group (kg = 0 or 2), N = lane&15
//   C 16x16:  VGPR v: lanes 0-15 -> M=m0+v, lanes 16-31 -> M=m0+v+8, N = lane&15
// ---------------------------------------------------------------------------
template <int KDIM, int NTN, int MT, bool RELU>
__global__ __launch_bounds__(256, 1)
void wmma_gemm_lds_kernel(const float* __restrict__ A,
                          const float* __restrict__ B,
                          const float* __restrict__ bias,
                          float*       __restrict__ C,
                          int M, int N, int mtiles) {
  constexpr int NB = NTN * 16;
  extern __shared__ float smem[];            // KDIM * NB floats, pair-interleaved

  const int tid   = (int)threadIdx.x;
  const int ncol0 = (int)blockIdx.x * NB;

  // Stage B panel: smem[(k>>1)*(2*NB) + col*2 + (k&1)] = B[k][ncol0+col]
  // (consecutive tid -> consecutive col -> coalesced global reads)
  for (int idx = tid; idx < KDIM * NB; idx += 256) {
    int k   = idx / NB;
    int col = idx - k * NB;
    smem[(k >> 1) * (2 * NB) + col * 2 + (k & 1)] =
        B[(size_t)k * N + ncol0 + col];
  }
  __syncthreads();

  const int wave = tid >> 5;
  const int lane = tid & 31;
  const int mt0  = ((int)blockIdx.y * 8 + wave) * MT;
  if (mt0 < mtiles) {                        // wave-uniform: EXEC all-ones
    const int kg   = (lane >> 4) << 1;       // 0 or 2
    const int lcol = lane & 15;

    const float* Arow[MT];
#pragma unroll
    for (int mi = 0; mi < MT; ++mi) {
      int mt_i = mt0 + mi; if (mt_i > mtiles - 1) mt_i = mtiles - 1; // clamp
      Arow[mi] = A + (size_t)(mt_i * 16 + (lane & 15)) * KDIM;
    }

    v8f acc[MT][NTN];
#pragma unroll
    for (int mi = 0; mi < MT; ++mi)
#pragma unroll
      for (int t = 0; t < NTN; ++t)
        acc[mi][t] = (v8f){0.f,0.f,0.f,0.f,0.f,0.f,0.f,0.f};

#pragma unroll 4
    for (int k0 = 0; k0 < KDIM; k0 += 4) {
      v2f a[MT];
#pragma unroll
      for (int mi = 0; mi < MT; ++mi)
        a[mi] = *(const v2f*)(Arow[mi] + k0 + kg);         // global, 8B

      const float* bbase = smem + ((k0 + kg) >> 1) * (2 * NB);
      v2f b[NTN];
#pragma unroll
      for (int t = 0; t < NTN; ++t)
        b[t] = *(const v2f*)(bbase + (t * 16 + lcol) * 2); // ds_load_b64

#pragma unroll
      for (int t = 0; t < NTN; ++t)          // t-outer: B pair consumers adjacent
#pragma unroll
        for (int mi = 0; mi < MT; ++mi)
          acc[mi][t] = __builtin_amdgcn_wmma_f32_16x16x4_f32(
              false, a[mi], false, b[t], (short)0, acc[mi][t], false, false);
    }

#pragma unroll
    for (int mi = 0; mi < MT; ++mi) {
      if (mt0 + mi < mtiles) {               // wave-uniform tail guard
        const int mbase = (mt0 + mi) * 16 + ((lane >> 4) << 3);
#pragma unroll
        for (int t = 0; t < NTN; ++t) {
          const int ncol = ncol0 + t * 16 + lcol;
          const float bv = bias[ncol];
#pragma unroll
          for (int v = 0; v < 8; ++v) {
            float r = acc[mi][t][v] + bv;
            if (RELU) r = fmaxf(r, 0.0f);
            C[(size_t)(mbase + v) * N + ncol] = r;
          }
        }
      }
    }
  }
}

// ---------------------------------------------------------------------------
// embed[b] = concat(x[head_idx[b]], rel_table[rel_ids[b]])  -> [BATCH, 2D]
// ---------------------------------------------------------------------------
__global__ void gather_embed_kernel(const float* __restrict__ x,
                                    const float* __restrict__ rel_table,
                                    const int*   __restrict__ head_idx,
                                    const int*   __restrict__ rel_ids,
                                    float*       __restrict__ embed) {
  int b = (int)blockIdx.x;
  int t = (int)threadIdx.x;                  // 0..127
  int h = head_idx[b];
  int r = rel_ids[b];
  embed[(size_t)b * (2 * DDIM) + t]        = x[(size_t)h * DDIM + t];
  embed[(size_t)b * (2 * DDIM) + DDIM + t] = rel_table[(size_t)r * DDIM + t];
}

// ---------------------------------------------------------------------------
extern "C" void kernel_launch(void* const* d_in, const int* in_sizes, int n_in,
                              void* d_out, int out_size, void* d_ws, size_t ws_size,
                              hipStream_t stream) {
  const float* entity_embed = (const float*)d_in[0];   // [N, D]
  const float* edge_val     = (const float*)d_in[1];   // [E]
  const float* gnn_W        = (const float*)d_in[2];   // [L, D, D]
  const float* gnn_b        = (const float*)d_in[3];   // [L, D]
  const float* rel_table    = (const float*)d_in[4];   // [R, D]
  const float* fc_W         = (const float*)d_in[5];   // [2D, NENT]
  const float* fc_b         = (const float*)d_in[6];   // [NENT]
  const int*   edge_row     = (const int*)d_in[7];     // [E]
  const int*   edge_col     = (const int*)d_in[8];     // [E]
  const int*   head_idx     = (const int*)d_in[9];     // [B]
  const int*   rel_ids      = (const int*)d_in[10];    // [B]
  float*       logits       = (float*)d_out;           // [B, NENT]

  const size_t feat_elems = (size_t)NNODES * DDIM;
  float* x_cur  = (float*)d_ws;
  float* x_next = x_cur  + feat_elems;
  float* agg    = x_next + feat_elems;
  float* embed  = agg    + feat_elems;                 // [B, 2D]

  // layer 0 input
  hipMemcpyAsync(x_cur, entity_embed, feat_elems * sizeof(float),
                 hipMemcpyDeviceToDevice, stream);

  const int spmm_blocks = (NEDGES * 32 + 255) / 256;   // 1 wave per edge

  // GNN dense layers: N = 128 -> one 8-tile panel (NB=128), 64 KB LDS.
  // MT=2: 16 WMMAs per k-step per wave; grid.y = ceil(3125 / 16) = 196.
  const int dense_mtiles = NNODES / 16;                // 3125
  dim3 dense_grid(1, (dense_mtiles + 15) / 16);        // (1, 196)
  const size_t dense_lds = (size_t)DDIM * 128 * sizeof(float);   // 64 KB

  for (int l = 0; l < NLAYERS; ++l) {
    hipMemsetAsync(agg, 0, feat_elems * sizeof(float), stream);

    spmm_edge_kernel<<<spmm_blocks, 256, 0, stream>>>(
        x_cur, edge_val, edge_row, edge_col, agg, NEDGES);

    wmma_gemm_lds_kernel<DDIM, 8, 2, true><<<dense_grid, 256, dense_lds, stream>>>(
        agg, gnn_W + (size_t)l * DDIM * DDIM, gnn_b + (size_t)l * DDIM,
        x_next, NNODES, DDIM, dense_mtiles);

    float* t = x_cur; x_cur = x_next; x_next = t;
  }

  gather_embed_kernel<<<BATCHSZ, DDIM, 0, stream>>>(
      x_cur, rel_table, head_idx, rel_ids, embed);

  // FC: N = 50000 = 625 panels of NB = 80 (NTN=5), K = 256 -> 80 KB LDS panel.
  // MT=2: each wave does 2 M-tiles x 5 N-tiles (10 WMMAs per k-step, B frags
  // reused across both M-tiles); grid.y = 64 / (8*2) = 4.
  dim3 fc_grid(NENT / 80, BATCHSZ / 16 / (8 * 2));     // (625, 4)
  const size_t fc_lds = (size_t)(2 * DDIM) * 80 * sizeof(float); // 80 KB

  wmma_gemm_lds_kernel<2 * DDIM, 5, 2, false><<<fc_grid, 256, fc_lds, stream>>>(
      embed, fc_W, fc_b, logits, BATCHSZ, NENT, BATCHSZ / 16);
}